// Net_31155692765518
// MI455X (gfx1250) — compile-verified
//
#include <hip/hip_runtime.h>
#include <hip/hip_bf16.h>

// ---------------------------------------------------------------------------
// GAT forward on MI455X (gfx1250). Edge-scatter bound; dense parts use fp32
// WMMA (V_WMMA_F32_16X16X4_F32) with LDS-staged A tiles so the K-loop is
// pure ds_load_b64 + global_load + v_wmma with no exec-mask control flow.
// ---------------------------------------------------------------------------

#define N_NODES 100000
#define N_EDGES 1600000
#define F_IN 65
#define HIDDEN 128
#define N_CLASSES 32
#define KPAD 68            // F_IN padded to multiple of 4
#define WPB 4              // waves per block in the node GEMM kernel

typedef __attribute__((ext_vector_type(2))) float v2f;
typedef __attribute__((ext_vector_type(8))) float v8f;

// Monotonic uint encoding of floats: enc(a) < enc(b) <=> a < b (for non-NaN).
__device__ __forceinline__ unsigned fenc(float f) {
    unsigned u = __float_as_uint(f);
    return (u & 0x80000000u) ? ~u : (u | 0x80000000u);
}
__device__ __forceinline__ float fdec(unsigned u) {
    u = (u & 0x80000000u) ? (u & 0x7FFFFFFFu) : ~u;
    return __uint_as_float(u);
}
#define ENC_NEG_INF 0x007FFFFFu  // fenc(-inf)

__device__ __forceinline__ float lrelu(float v) { return v > 0.f ? v : 0.2f * v; }

// ---------------------------------------------------------------------------
// Pass 0: init accumulators
// ---------------------------------------------------------------------------
__global__ void init_kernel(float* __restrict__ out1, float* __restrict__ out2,
                            float* __restrict__ den1, float* __restrict__ den2,
                            unsigned* __restrict__ m1, unsigned* __restrict__ m2) {
    int t = blockIdx.x * blockDim.x + threadIdx.x;
    if (t < N_NODES * N_CLASSES) { out1[t] = 0.f; out2[t] = 0.f; }
    if (t < N_NODES) {
        den1[t] = 0.f; den2[t] = 0.f;
        m1[t] = ENC_NEG_INF; m2[t] = ENC_NEG_INF;
    }
}

// ---------------------------------------------------------------------------
// Fused node GEMMs: per 16-row tile (one wave):
//   x0 = relu(x @ W_mlp + b)   [16,65]x[65,128]  (A staged in LDS, K pad 68)
//   h1 = x0 @ W1, h2 = x0 @ W2 [16,128]x[128,32] (x0 tile kept in LDS)
// f32 WMMA layouts:
//   A 16x4: lanes 0-15 K=k,k+1 ; lanes 16-31 K=k+2,k+3 ; M=lane&15
//   B 4x16: same K split; N=lane&15
//   C 16x16: VGPR r -> M=r (+8 for hi lanes), N=lane&15
// B tail for GEMM1 is branchless: padded A rows are zero for k>=65, so the
// B row index is simply clamped (product is zero anyway).
// ---------------------------------------------------------------------------
__global__ __launch_bounds__(WPB * 32) void node_gemm_kernel(
        const float* __restrict__ x, const float* __restrict__ Wm,
        const float* __restrict__ bias,
        const float* __restrict__ W1, const float* __restrict__ W2,
        float* __restrict__ x0, float* __restrict__ h1, float* __restrict__ h2) {
    __shared__ float lds[WPB * (16 * KPAD + 16 * HIDDEN)];   // 50 KB
    const int wave = threadIdx.x >> 5;
    const int lane = threadIdx.x & 31;
    const int tile = blockIdx.x * WPB + wave;
    if (tile * 16 >= N_NODES) return;        // wave-uniform: EXEC stays all-ones
    const int r0 = tile * 16;

    float* aLds  = lds + wave * (16 * KPAD + 16 * HIDDEN);
    float* x0Lds = aLds + 16 * KPAD;

    // Stage A tile (16 x 65 -> 16 x 68 zero-padded), coalesced by the wave.
    for (int i = lane; i < 16 * KPAD; i += 32) {
        const int r = i / KPAD, k = i % KPAD;
        aLds[i] = (k < F_IN) ? x[(size_t)(r0 + r) * F_IN + k] : 0.f;
    }
    asm volatile("s_wait_dscnt 0" ::: "memory");

    const int mrow  = lane & 15;
    const int ksub  = (lane >> 4) * 2;       // 0 or 2
    const int mbase = (lane < 16) ? 0 : 8;

    // ---- GEMM1: x0 = relu(x @ Wm + b) ----
    for (int nt = 0; nt < HIDDEN / 16; ++nt) {
        const int n = nt * 16 + mrow;
        v8f acc = {};
        #pragma unroll
        for (int k = 0; k < KPAD; k += 4) {
            const int ka = k + ksub;
            const v2f a = *(const v2f*)&aLds[mrow * KPAD + ka];   // ds_load_b64
            v2f b;
            b.x = Wm[(size_t)min(ka,     F_IN - 1) * HIDDEN + n]; // clamped, branchless
            b.y = Wm[(size_t)min(ka + 1, F_IN - 1) * HIDDEN + n];
            acc = __builtin_amdgcn_wmma_f32_16x16x4_f32(
                false, a, false, b, (short)0, acc, false, false);
        }
        const float bv = bias[n];
        #pragma unroll
        for (int r = 0; r < 8; ++r) {
            const float v = fmaxf(acc[r] + bv, 0.f);
            x0[(size_t)(r0 + mbase + r) * HIDDEN + n] = v;
            x0Lds[(mbase + r) * HIDDEN + n] = v;
        }
    }
    asm volatile("s_wait_dscnt 0" ::: "memory");

    // ---- GEMM2: h1 = x0 @ W1, h2 = x0 @ W2 (x0 tile read from LDS) ----
    const float* Ws[2] = {W1, W2};
    float* Hs[2] = {h1, h2};
    for (int w = 0; w < 2; ++w) {
        const float* __restrict__ W = Ws[w];
        float* __restrict__ H = Hs[w];
        for (int nt = 0; nt < N_CLASSES / 16; ++nt) {
            const int n = nt * 16 + mrow;
            v8f acc = {};
            #pragma unroll
            for (int k = 0; k < HIDDEN; k += 4) {
                const int ka = k + ksub;
                const v2f a = *(const v2f*)&x0Lds[mrow * HIDDEN + ka];
                v2f b;
                b.x = W[(ka    ) * N_CLASSES + n];
                b.y = W[(ka + 1) * N_CLASSES + n];
                acc = __builtin_amdgcn_wmma_f32_16x16x4_f32(
                    false, a, false, b, (short)0, acc, false, false);
            }
            #pragma unroll
            for (int r = 0; r < 8; ++r)
                H[(size_t)(r0 + mbase + r) * N_CLASSES + n] = acc[r];
        }
    }
}

// ---------------------------------------------------------------------------
// Per-node attention scalars s = h@a_src, d = h@a_dst (both convs)
// ---------------------------------------------------------------------------
__global__ void sd_kernel(const float* __restrict__ h1, const float* __restrict__ h2,
                          const float* __restrict__ a1s, const float* __restrict__ a1d,
                          const float* __restrict__ a2s, const float* __restrict__ a2d,
                          float* __restrict__ s1, float* __restrict__ d1,
                          float* __restrict__ s2, float* __restrict__ d2) {
    int n = blockIdx.x * blockDim.x + threadIdx.x;
    if (n >= N_NODES) return;
    float S1 = 0.f, D1 = 0.f, S2 = 0.f, D2 = 0.f;
    #pragma unroll
    for (int c = 0; c < N_CLASSES; ++c) {
        const float v1 = h1[n * N_CLASSES + c];
        const float v2 = h2[n * N_CLASSES + c];
        S1 += v1 * a1s[c]; D1 += v1 * a1d[c];
        S2 += v2 * a2s[c]; D2 += v2 * a2d[c];
    }
    s1[n] = S1; d1[n] = D1; s2[n] = S2; d2[n] = D2;
}

// ---------------------------------------------------------------------------
// Edge pass 1: e = leaky_relu(s[src]+d[dst]); segment-max via encoded atomicMax
// ---------------------------------------------------------------------------
__global__ void edge_pass1(const int* __restrict__ src, const int* __restrict__ dst,
                           const float* __restrict__ s1, const float* __restrict__ d1,
                           const float* __restrict__ s2, const float* __restrict__ d2,
                           float* __restrict__ e1, float* __restrict__ e2,
                           unsigned* __restrict__ m1, unsigned* __restrict__ m2) {
    int e = blockIdx.x * blockDim.x + threadIdx.x;
    if (e >= N_EDGES) return;
    const int u = src[e], v = dst[e];
    const float v1 = lrelu(s1[u] + d1[v]);
    const float v2 = lrelu(s2[u] + d2[v]);
    e1[e] = v1; e2[e] = v2;
    atomicMax(&m1[v], fenc(v1));
    atomicMax(&m2[v], fenc(v2));
}

// ---------------------------------------------------------------------------
// Edge pass 2: ex = exp(e - m_hat[dst]); denom atomicAdd (folds isfinite fix)
// ---------------------------------------------------------------------------
__global__ void edge_pass2(const int* __restrict__ dst,
                           float* __restrict__ e1, float* __restrict__ e2,
                           const unsigned* __restrict__ m1, const unsigned* __restrict__ m2,
                           float* __restrict__ den1, float* __restrict__ den2) {
    int e = blockIdx.x * blockDim.x + threadIdx.x;
    if (e >= N_EDGES) return;
    const int v = dst[e];
    float mm1 = fdec(m1[v]); if (!isfinite(mm1)) mm1 = 0.f;
    float mm2 = fdec(m2[v]); if (!isfinite(mm2)) mm2 = 0.f;
    const float ex1 = __expf(e1[e] - mm1);
    const float ex2 = __expf(e2[e] - mm2);
    e1[e] = ex1; e2[e] = ex2;
    atomicAdd(&den1[v], ex1);
    atomicAdd(&den2[v], ex2);
}

// ---------------------------------------------------------------------------
// Edge pass 3: one wave32 per edge, lane == class (N_CLASSES == 32):
// coalesced gather of h[src] row + coalesced 32-float atomic scatter to dst.
// ---------------------------------------------------------------------------
__global__ __launch_bounds__(256) void edge_pass3(
        const int* __restrict__ src, const int* __restrict__ dst,
        const float* __restrict__ e1, const float* __restrict__ e2,
        const float* __restrict__ den1, const float* __restrict__ den2,
        const float* __restrict__ h1, const float* __restrict__ h2,
        float* __restrict__ out1, float* __restrict__ out2) {
    const int wave = (blockIdx.x * blockDim.x + threadIdx.x) >> 5;
    const int c = threadIdx.x & 31;
    if (wave >= N_EDGES) return;
    const int u = src[wave], v = dst[wave];
    const float a1 = e1[wave] / (den1[v] + 1e-16f);
    const float a2 = e2[wave] / (den2[v] + 1e-16f);
    atomicAdd(&out1[v * N_CLASSES + c], h1[u * N_CLASSES + c] * a1);
    atomicAdd(&out2[v * N_CLASSES + c], h2[u * N_CLASSES + c] * a2);
}

// ---------------------------------------------------------------------------
// Finalize: one wave per node: x3 = rowmax(x0); x1 = relu(out1+b1);
// x2 = out2+b2; logits = concat + x; log_softmax over 65 via wave shuffles.
// ---------------------------------------------------------------------------
__global__ __launch_bounds__(256) void finalize_kernel(
        const float* __restrict__ x, const float* __restrict__ x0,
        const float* __restrict__ out1, const float* __restrict__ out2,
        const float* __restrict__ b1, const float* __restrict__ b2,
        float* __restrict__ out) {
    const int node = (blockIdx.x * blockDim.x + threadIdx.x) >> 5;
    const int lane = threadIdx.x & 31;
    if (node >= N_NODES) return;

    const float* xr = x0 + (size_t)node * HIDDEN;
    float mx = fmaxf(fmaxf(xr[lane], xr[lane + 32]), fmaxf(xr[lane + 64], xr[lane + 96]));
    #pragma unroll
    for (int off = 16; off; off >>= 1) mx = fmaxf(mx, __shfl_xor(mx, off, 32));

    const float* xin = x + (size_t)node * F_IN;
    const float l0 = fmaxf(out1[node * N_CLASSES + lane] + b1[lane], 0.f) + xin[lane];
    const float l1 = (out2[node * N_CLASSES + lane] + b2[lane]) + xin[lane + 32];
    const float l2 = mx + xin[64];  // identical across lanes (mx reduced)

    float lm = fmaxf(fmaxf(l0, l1), l2);
    #pragma unroll
    for (int off = 16; off; off >>= 1) lm = fmaxf(lm, __shfl_xor(lm, off, 32));

    float s = __expf(l0 - lm) + __expf(l1 - lm) + ((lane == 0) ? __expf(l2 - lm) : 0.f);
    #pragma unroll
    for (int off = 16; off; off >>= 1) s += __shfl_xor(s, off, 32);
    const float lse = lm + __logf(s);

    float* orow = out + (size_t)node * F_IN;
    orow[lane]      = l0 - lse;
    orow[lane + 32] = l1 - lse;
    if (lane == 0) orow[64] = l2 - lse;
}

// ---------------------------------------------------------------------------
extern "C" void kernel_launch(void* const* d_in, const int* in_sizes, int n_in,
                              void* d_out, int out_size, void* d_ws, size_t ws_size,
                              hipStream_t stream) {
    const float* x     = (const float*)d_in[0];
    const int*   ei    = (const int*)d_in[1];
    const float* W_mlp = (const float*)d_in[2];
    const float* b_mlp = (const float*)d_in[3];
    const float* W1    = (const float*)d_in[4];
    const float* a1s   = (const float*)d_in[5];
    const float* a1d   = (const float*)d_in[6];
    const float* b1    = (const float*)d_in[7];
    const float* W2    = (const float*)d_in[8];
    const float* a2s   = (const float*)d_in[9];
    const float* a2d   = (const float*)d_in[10];
    const float* b2    = (const float*)d_in[11];
    const int* src = ei;
    const int* dst = ei + N_EDGES;
    float* out = (float*)d_out;

    // workspace layout (floats): ~118 MB total
    float* ws = (float*)d_ws;
    size_t off = 0;
    float* x0   = ws + off; off += (size_t)N_NODES * HIDDEN;
    float* h1   = ws + off; off += (size_t)N_NODES * N_CLASSES;
    float* h2   = ws + off; off += (size_t)N_NODES * N_CLASSES;
    float* s1   = ws + off; off += N_NODES;
    float* d1   = ws + off; off += N_NODES;
    float* s2   = ws + off; off += N_NODES;
    float* d2   = ws + off; off += N_NODES;
    unsigned* m1 = (unsigned*)(ws + off); off += N_NODES;
    unsigned* m2 = (unsigned*)(ws + off); off += N_NODES;
    float* den1 = ws + off; off += N_NODES;
    float* den2 = ws + off; off += N_NODES;
    float* e1   = ws + off; off += N_EDGES;
    float* e2   = ws + off; off += N_EDGES;
    float* out1 = ws + off; off += (size_t)N_NODES * N_CLASSES;
    float* out2 = ws + off; off += (size_t)N_NODES * N_CLASSES;

    const int TB = 256;
    const int n_tiles = N_NODES / 16;  // 6250
    init_kernel<<<(N_NODES * N_CLASSES + TB - 1) / TB, TB, 0, stream>>>(
        out1, out2, den1, den2, m1, m2);
    node_gemm_kernel<<<(n_tiles + WPB - 1) / WPB, WPB * 32, 0, stream>>>(
        x, W_mlp, b_mlp, W1, W2, x0, h1, h2);
    sd_kernel<<<(N_NODES + TB - 1) / TB, TB, 0, stream>>>(
        h1, h2, a1s, a1d, a2s, a2d, s1, d1, s2, d2);
    edge_pass1<<<(N_EDGES + TB - 1) / TB, TB, 0, stream>>>(
        src, dst, s1, d1, s2, d2, e1, e2, m1, m2);
    edge_pass2<<<(N_EDGES + TB - 1) / TB, TB, 0, stream>>>(
        dst, e1, e2, m1, m2, den1, den2);
    edge_pass3<<<((size_t)N_EDGES * 32 + TB - 1) / TB, TB, 0, stream>>>(
        src, dst, e1, e2, den1, den2, h1, h2, out1, out2);
    finalize_kernel<<<(N_NODES * 32 + TB - 1) / TB, TB, 0, stream>>>(
        x, x0, out1, out2, b1, b2, out);
}